// GroupPoolSet2Set_4088808866416
// MI455X (gfx1250) — compile-verified
//
#include <hip/hip_runtime.h>
#include <math.h>

#define D 128
#define CHUNK 256

typedef __attribute__((ext_vector_type(2))) float v2f;
typedef __attribute__((ext_vector_type(4))) float v4f;
typedef __attribute__((ext_vector_type(8))) float v8f;

__device__ __forceinline__ float sigm(float x) { return 1.0f / (1.0f + expf(-x)); }

// order-preserving float <-> uint encoding for atomicMax-based segment max
__device__ __forceinline__ unsigned fenc(float f) {
    unsigned u = __float_as_uint(f);
    return (u & 0x80000000u) ? ~u : (u | 0x80000000u);
}
__device__ __forceinline__ float fdec(unsigned u) {
    unsigned b = (u & 0x80000000u) ? (u & 0x7FFFFFFFu) : ~u;
    return __uint_as_float(b);
}

__global__ void k_zero_u32(unsigned* p, int n) {
    int i = blockIdx.x * blockDim.x + threadIdx.x;
    if (i < n) p[i] = 0u;
}
__global__ void k_fill_i32(int* p, int v, int n) {
    int i = blockIdx.x * blockDim.x + threadIdx.x;
    if (i < n) p[i] = v;
}
__global__ void k_bincount(const int* __restrict__ bs, int* bc, int n) {
    int i = blockIdx.x * blockDim.x + threadIdx.x;
    if (i < n) atomicAdd(&bc[bs[i]], 1);
}
// first-index argmax of bc[0..Bn)
__global__ void k_argmax(const int* __restrict__ bc, int* defb, int Bn) {
    __shared__ int sc[256], si[256];
    int t = threadIdx.x;
    int bestc = -1, besti = 0;
    for (int j = t; j < Bn; j += 256) {
        int c = bc[j];
        if (c > bestc) { bestc = c; besti = j; }
    }
    sc[t] = bestc; si[t] = besti;
    __syncthreads();
    for (int off = 128; off > 0; off >>= 1) {
        if (t < off) {
            if (sc[t + off] > sc[t] || (sc[t + off] == sc[t] && si[t + off] < si[t])) {
                sc[t] = sc[t + off]; si[t] = si[t + off];
            }
        }
        __syncthreads();
    }
    if (t == 0) defb[0] = si[0];
}
__global__ void k_scatter(const int* __restrict__ a2g0, const int* __restrict__ a2g1,
                          const int* __restrict__ bs, int* gb, int n) {
    int i = blockIdx.x * blockDim.x + threadIdx.x;
    if (i < n) gb[a2g0[i]] = bs[a2g1[i]];
}
__global__ void k_fix(int* gb, const int* __restrict__ defb, int* counts, int n) {
    int i = blockIdx.x * blockDim.x + threadIdx.x;
    if (i >= n) return;
    int b = gb[i];
    if (b < 0) { b = *defb; gb[i] = b; }
    atomicAdd(&counts[b], 1);
}

// Step-1 LSTM (h=c=q*=0 -> single vector) + fold constants into base[4D].
// base[j] = b_ih[j]+b_hh[j] + sum_k h1[k]*(W_ih[j,k] + W_hh[j,k])
__global__ void k_lstm1(const float* __restrict__ Wih, const float* __restrict__ Whh,
                        const float* __restrict__ bih, const float* __restrict__ bhh,
                        float* h1, float* c1, float* base) {
    __shared__ float h1s[D];
    int t = threadIdx.x;  // 0..511
    if (t < D) {
        float iv = bih[t] + bhh[t];
        float gv = bih[2 * D + t] + bhh[2 * D + t];
        float ov = bih[3 * D + t] + bhh[3 * D + t];
        float c = sigm(iv) * tanhf(gv);       // sig(f)*c_prev = 0
        float h = sigm(ov) * tanhf(c);
        c1[t] = c; h1[t] = h; h1s[t] = h;
    }
    __syncthreads();
    float s = bih[t] + bhh[t];
    for (int k = 0; k < D; k++)
        s += h1s[k] * (Wih[t * (2 * D) + k] + Whh[t * D + k]);
    base[t] = s;
}

// one wave per row: e[i] = x_i . q   (q broadcast or per-batch); segment max via atomicMax
// x is streamed non-temporally (512MB, no reuse -> don't thrash L2).
__global__ void k_dotmax(const float* __restrict__ x, const float* __restrict__ q,
                         const int* __restrict__ gb, float* e, unsigned* emax,
                         int n, int perBatch) {
    int wid = (blockIdx.x * blockDim.x + threadIdx.x) >> 5;
    int lane = threadIdx.x & 31;
    if (wid >= n) return;
    int b = gb[wid];
    const v4f* xr = (const v4f*)(x + (size_t)wid * D);
    const v4f* qr = (const v4f*)(q + (perBatch ? (size_t)b * D : 0));
    v4f xv = __builtin_nontemporal_load(&xr[lane]);
    v4f qv = qr[lane];
    float s = xv.x * qv.x + xv.y * qv.y + xv.z * qv.z + xv.w * qv.w;
    for (int m = 16; m >= 1; m >>= 1) s += __shfl_xor(s, m, 32);
    if (lane == 0) {
        e[wid] = s;
        atomicMax(&emax[b], fenc(s));
    }
}

// ex = exp(e - emax[gb]) in place; denom segment-sum
__global__ void k_expsum(float* e, const int* __restrict__ gb,
                         const unsigned* __restrict__ emax, float* denom, int n) {
    int i = blockIdx.x * blockDim.x + threadIdx.x;
    if (i >= n) return;
    int b = gb[i];
    float v = expf(e[i] - fdec(emax[b]));
    e[i] = v;
    atomicAdd(&denom[b], v);
}

// r[b] += (ex/denom[b]) * x_row ; thread t owns dim t; register accum, flush on batch change.
// Works for arbitrary gb ordering; near-zero atomics when gb is sorted (it is here).
__global__ void k_raccum(const float* __restrict__ x, const float* __restrict__ ex,
                         const int* __restrict__ gb, const float* __restrict__ denom,
                         float* r, int n) {
    int t = threadIdx.x;  // 0..127
    int start = blockIdx.x * CHUNK;
    int end = start + CHUNK; if (end > n) end = n;
    float acc = 0.0f, dinv = 0.0f;
    int cur = -1;
    for (int row = start; row < end; ++row) {
        int b = gb[row];
        if (b != cur) {
            if (cur >= 0) atomicAdd(&r[(size_t)cur * D + t], acc);
            acc = 0.0f; cur = b; dinv = 1.0f / denom[b];
        }
        float a = ex[row] * dinv;
        float xv = __builtin_nontemporal_load(&x[(size_t)row * D + t]);
        acc = fmaf(a, xv, acc);
    }
    if (cur >= 0) atomicAdd(&r[(size_t)cur * D + t], acc);
}

// Step-2 gates GEMM on the matrix pipe (f32 WMMA 16x16x4) fused with LSTM cell:
//   gates2[b,j] = base[j] + sum_k r1[b,k] * W_ih[j, 128+k]
// One wave computes a 16(M)x16(T) tile for all four gates, then h2 = sig(o)*tanh(c2).
__global__ void k_gates_wmma(const float* __restrict__ r1, const float* __restrict__ Wih,
                             const float* __restrict__ base, const float* __restrict__ c1,
                             float* __restrict__ h2, int Bn) {
    int wave = (blockIdx.x * blockDim.x + threadIdx.x) >> 5;
    int lane = threadIdx.x & 31;
    const int tTiles = D / 16;                  // 8
    int m0 = (wave / tTiles) * 16;
    int t0 = (wave % tTiles) * 16;
    if (m0 >= Bn) return;

    int kh = (lane < 16) ? 0 : 2;               // K sub-offset per lane half (ISA A/B layout)
    int lr = lane & 15;

    const float* Arow = r1 + (size_t)(m0 + lr) * D + kh;
    // B[k][j] = W_ih[j*(2D) + D + k], columns j = gate*D + t0 + lr
    const float* Bi = Wih + (size_t)(0 * D + t0 + lr) * (2 * D) + D + kh;
    const float* Bf = Wih + (size_t)(1 * D + t0 + lr) * (2 * D) + D + kh;
    const float* Bg = Wih + (size_t)(2 * D + t0 + lr) * (2 * D) + D + kh;
    const float* Bo = Wih + (size_t)(3 * D + t0 + lr) * (2 * D) + D + kh;

    v8f ai = {}, af = {}, ag = {}, ao = {};
    #pragma unroll 4
    for (int k = 0; k < D; k += 4) {
        v2f a; a.x = Arow[k]; a.y = Arow[k + 1];
        v2f bi; bi.x = Bi[k]; bi.y = Bi[k + 1];
        v2f bf; bf.x = Bf[k]; bf.y = Bf[k + 1];
        v2f bg; bg.x = Bg[k]; bg.y = Bg[k + 1];
        v2f bo; bo.x = Bo[k]; bo.y = Bo[k + 1];
        ai = __builtin_amdgcn_wmma_f32_16x16x4_f32(false, a, false, bi, (short)0, ai, false, false);
        af = __builtin_amdgcn_wmma_f32_16x16x4_f32(false, a, false, bf, (short)0, af, false, false);
        ag = __builtin_amdgcn_wmma_f32_16x16x4_f32(false, a, false, bg, (short)0, ag, false, false);
        ao = __builtin_amdgcn_wmma_f32_16x16x4_f32(false, a, false, bo, (short)0, ao, false, false);
    }

    int t = t0 + lr;
    float b_i = base[t], b_f = base[D + t], b_g = base[2 * D + t], b_o = base[3 * D + t];
    float cp = c1[t];
    int rbase = m0 + ((lane < 16) ? 0 : 8);
    #pragma unroll
    for (int rr = 0; rr < 8; ++rr) {
        int row = rbase + rr;
        float iv = ai[rr] + b_i;
        float fv = af[rr] + b_f;
        float gv = ag[rr] + b_g;
        float ov = ao[rr] + b_o;
        float c2 = sigm(fv) * cp + sigm(iv) * tanhf(gv);
        h2[(size_t)row * D + t] = sigm(ov) * tanhf(c2);
    }
}

__global__ void k_output(const float* __restrict__ h2, const float* __restrict__ r2,
                         const int* __restrict__ counts, float* out, int total) {
    int i = blockIdx.x * blockDim.x + threadIdx.x;
    if (i >= total) return;
    int b = i >> 8;           // / (2*D)
    int t = i & 255;
    float v = 0.0f;
    if (counts[b] > 0)
        v = (t < D) ? h2[(size_t)b * D + t] : r2[(size_t)b * D + (t - D)];
    out[i] = v;
}

static inline int cdiv(long long a, long long b) { return (int)((a + b - 1) / b); }

extern "C" void kernel_launch(void* const* d_in, const int* in_sizes, int n_in,
                              void* d_out, int out_size, void* d_ws, size_t ws_size,
                              hipStream_t stream) {
    (void)n_in; (void)ws_size;
    const float* xg  = (const float*)d_in[0];
    const int*   a2g = (const int*)d_in[1];
    const int*   bs  = (const int*)d_in[2];
    const float* Wih = (const float*)d_in[3];
    const float* Whh = (const float*)d_in[4];
    const float* bih = (const float*)d_in[5];
    const float* bhh = (const float*)d_in[6];
    float* out = (float*)d_out;

    const int N  = in_sizes[0] / D;       // 1,000,000
    const int Bn = out_size / (2 * D);    // 4096

    // workspace carve-out (256B aligned)
    char* w = (char*)d_ws;
    auto carve = [&](size_t bytes) -> char* {
        char* p = w; w += (bytes + 255) & ~(size_t)255; return p;
    };
    float*    r1     = (float*)carve((size_t)Bn * D * 4);
    float*    r2     = (float*)carve((size_t)Bn * D * 4);
    unsigned* stats  = (unsigned*)carve((size_t)Bn * 6 * 4);   // bc|counts|emax1|denom1|emax2|denom2
    int*      bc     = (int*)stats;
    int*      counts = (int*)(stats + Bn);
    unsigned* emax1  = stats + 2 * Bn;
    float*    denom1 = (float*)(stats + 3 * Bn);
    unsigned* emax2  = stats + 4 * Bn;
    float*    denom2 = (float*)(stats + 5 * Bn);
    float*    h1     = (float*)carve(D * 4);
    float*    c1     = (float*)carve(D * 4);
    float*    base   = (float*)carve(4 * D * 4);
    int*      defb   = (int*)carve(4);
    float*    h2     = (float*)carve((size_t)Bn * D * 4);
    int*      gb     = (int*)carve((size_t)N * 4);
    float*    e      = (float*)carve((size_t)N * 4);

    const int TPB = 256;

    // init accumulators (r1,r2 contiguous; stats contiguous)
    k_zero_u32<<<cdiv((long long)Bn * 2 * D, TPB), TPB, 0, stream>>>((unsigned*)r1, Bn * 2 * D);
    k_zero_u32<<<cdiv((long long)Bn * 6, TPB), TPB, 0, stream>>>(stats, Bn * 6);
    k_fill_i32<<<cdiv(N, TPB), TPB, 0, stream>>>(gb, -1, N);

    // group -> batch assignment
    k_bincount<<<cdiv(N, TPB), TPB, 0, stream>>>(bs, bc, N);
    k_argmax<<<1, 256, 0, stream>>>(bc, defb, Bn);
    k_scatter<<<cdiv(N, TPB), TPB, 0, stream>>>(a2g, a2g + N, bs, gb, N);
    k_fix<<<cdiv(N, TPB), TPB, 0, stream>>>(gb, defb, counts, N);

    // step 1: LSTM collapses to one vector; fold constants into base
    k_lstm1<<<1, 4 * D, 0, stream>>>(Wih, Whh, bih, bhh, h1, c1, base);
    k_dotmax<<<cdiv((long long)N * 32, TPB), TPB, 0, stream>>>(xg, h1, gb, e, emax1, N, 0);
    k_expsum<<<cdiv(N, TPB), TPB, 0, stream>>>(e, gb, emax1, denom1, N);
    k_raccum<<<cdiv(N, CHUNK), D, 0, stream>>>(xg, e, gb, denom1, r1, N);

    // step 2: WMMA gates GEMM + fused cell -> h2, then attention again
    {
        long long waves = (long long)(Bn / 16) * (D / 16);
        k_gates_wmma<<<cdiv(waves * 32, TPB), TPB, 0, stream>>>(r1, Wih, base, c1, h2, Bn);
    }
    k_dotmax<<<cdiv((long long)N * 32, TPB), TPB, 0, stream>>>(xg, h2, gb, e, emax2, N, 1);
    k_expsum<<<cdiv(N, TPB), TPB, 0, stream>>>(e, gb, emax2, denom2, N);
    k_raccum<<<cdiv(N, CHUNK), D, 0, stream>>>(xg, e, gb, denom2, r2, N);

    // output: [h2 | r2] masked by presence
    k_output<<<cdiv((long long)Bn * 2 * D, TPB), TPB, 0, stream>>>(h2, r2, counts, out, Bn * 2 * D);
}